// PhaseWeightedAttention_55439437857257
// MI455X (gfx1250) — compile-verified
//
#include <hip/hip_runtime.h>
#include <math.h>

#define B_   4
#define N_   4096
#define D_   1024
#define EPS_ 1e-6f
#define NC_  8              // N-chunks per (b, d-tile) in the WMMA reduction
#define NT_  (D_ / 16)      // 64 d-tiles of 16 columns

typedef float v2f __attribute__((ext_vector_type(2)));
typedef float v8f __attribute__((ext_vector_type(8)));

// ---------------- zero accumulators (CS + sums are contiguous in ws) ----------------
__global__ __launch_bounds__(256) void pwa_zero(float* __restrict__ p, int n) {
    int i = blockIdx.x * 256 + threadIdx.x;
    if (i < n) p[i] = 0.0f;
}

// ------- cos/sin interleaved table + per-batch C = sum(cos), S = sum(sin) -------
__global__ __launch_bounds__(256) void pwa_phase(const float* __restrict__ phases,
                                                 float* __restrict__ cs,     // [B*N][2]
                                                 float* __restrict__ CS) {
    __shared__ float sc[256], ss[256];
    int i = blockIdx.x * 256 + threadIdx.x;   // grid exactly covers B*N
    int b = i / N_;                           // 256 | N_, so whole block is one batch
    float ph = phases[i];
    float c = cosf(ph), s = sinf(ph);
    float2 pr; pr.x = c; pr.y = s;
    *(float2*)(cs + 2 * i) = pr;
    sc[threadIdx.x] = c;
    ss[threadIdx.x] = s;
    __syncthreads();
    for (int off = 128; off > 0; off >>= 1) {
        if (threadIdx.x < off) {
            sc[threadIdx.x] += sc[threadIdx.x + off];
            ss[threadIdx.x] += ss[threadIdx.x + off];
        }
        __syncthreads();
    }
    if (threadIdx.x == 0) {
        atomicAdd(&CS[b * 2 + 0], sc[0]);
        atomicAdd(&CS[b * 2 + 1], ss[0]);
    }
}

// ---------------- rank-3 reduction via V_WMMA_F32_16X16X4_F32 ----------------
// D tile rows: m=0 -> H[d] = sum_n hidden, m=1 -> HC[d] = sum_n cos*hidden,
//              m=2 -> HS[d] = sum_n sin*hidden. One wave per (b, d-tile, chunk).
__global__ __launch_bounds__(32) void pwa_wmma_reduce(const float* __restrict__ hidden,
                                                      const float* __restrict__ cs,
                                                      float* __restrict__ sums) {
    const int lane = threadIdx.x;             // wave32, EXEC all ones
    int blk = blockIdx.x;
    int b   = blk / (NT_ * NC_);
    int rem = blk % (NT_ * NC_);
    int dt  = rem / NC_;
    int ch  = rem % NC_;
    const int d0 = dt * 16;
    const int n0 = ch * (N_ / NC_);
    const int m  = lane & 15;                 // A-matrix row / B-matrix column
    const int kb = (lane >> 4) << 1;          // lanes 0-15: K=0,1 ; lanes 16-31: K=2,3

    // branch-free row-selection masks, hoisted out of the loop
    const float f0 = (m == 0) ? 1.0f : 0.0f;
    const float f1 = (m == 1) ? 1.0f : 0.0f;
    const float f2 = (m == 2) ? 1.0f : 0.0f;

    v8f acc = {};
    const float* hb = hidden + (size_t)b * N_ * D_ + d0 + m;
    const float* cb = cs + (size_t)(b * N_) * 2;

    for (int n = n0; n < n0 + (N_ / NC_); n += 4) {
        // one b128 load: (c0,s0,c1,s1) at rows n+kb, n+kb+1
        const float4 p = *(const float4*)(cb + 2 * (n + kb));
        v2f A;
        A.x = f0 + f1 * p.x + f2 * p.y;
        A.y = f0 + f1 * p.z + f2 * p.w;

        const float* hp = hb + (size_t)(n + kb) * D_;
        v2f Bv; Bv.x = hp[0]; Bv.y = hp[D_];

        acc = __builtin_amdgcn_wmma_f32_16x16x4_f32(
            /*neg_a=*/false, A, /*neg_b=*/false, Bv,
            /*c_mod=*/(short)0, acc, /*reuse_a=*/false, /*reuse_b=*/false);
    }

    if (lane < 16) {                          // rows 0..2 hold H/HC/HS partials
        float* sb = sums + (size_t)b * 3 * D_ + d0 + lane;
        atomicAdd(sb,          acc[0]);
        atomicAdd(sb + D_,     acc[1]);
        atomicAdd(sb + 2 * D_, acc[2]);
    }
}

// ---------------- fused normalize + blend, streaming float4 ----------------
__global__ __launch_bounds__(256) void pwa_finalize(const float* __restrict__ hidden,
                                                    const float* __restrict__ cs,
                                                    const float* __restrict__ CS,
                                                    const float* __restrict__ sums,
                                                    const float* __restrict__ alpha_p,
                                                    float* __restrict__ out) {
    __shared__ float s3[3 * D_];              // H | HC | HS for this batch (12 KB)
    const int tid = threadIdx.x;
    const int ROWS = 16;
    int b  = blockIdx.x / (N_ / ROWS);
    int n0 = (blockIdx.x % (N_ / ROWS)) * ROWS;

    const float* sp = sums + (size_t)b * 3 * D_;
    for (int j = tid; j < 3 * D_; j += 256) s3[j] = sp[j];
    __syncthreads();

    float a = alpha_p[0];
    a = fminf(fmaxf(a, 0.0f), 1.0f);
    float om = 1.0f - a;
    float Cb = CS[b * 2 + 0], Sb = CS[b * 2 + 1];

    int d = tid * 4;                          // 256 threads * float4 = full row of 1024
    for (int r = 0; r < ROWS; ++r) {
        int n = n0 + r;
        const float2 pr = *(const float2*)(cs + 2 * (b * N_ + n));
        float cn = pr.x, sn = pr.y;
        float t  = 0.5f * ((float)N_ + cn * Cb + sn * Sb);
        float sc = a * 0.5f / fmaxf(t, EPS_);

        const float4 h = *(const float4*)(hidden + ((size_t)(b * N_ + n)) * D_ + d);
        float4 o;
        o.x = sc * (s3[d+0] + cn * s3[D_+d+0] + sn * s3[2*D_+d+0]) + om * h.x;
        o.y = sc * (s3[d+1] + cn * s3[D_+d+1] + sn * s3[2*D_+d+1]) + om * h.y;
        o.z = sc * (s3[d+2] + cn * s3[D_+d+2] + sn * s3[2*D_+d+2]) + om * h.z;
        o.w = sc * (s3[d+3] + cn * s3[D_+d+3] + sn * s3[2*D_+d+3]) + om * h.w;
        *(float4*)(out + ((size_t)(b * N_ + n)) * D_ + d) = o;
    }
}

extern "C" void kernel_launch(void* const* d_in, const int* in_sizes, int n_in,
                              void* d_out, int out_size, void* d_ws, size_t ws_size,
                              hipStream_t stream) {
    const float* hidden = (const float*)d_in[0];   // [B, N, D] f32
    const float* phases = (const float*)d_in[1];   // [B, N]    f32
    const float* alpha  = (const float*)d_in[2];   // scalar    f32
    float* out = (float*)d_out;

    // workspace layout: cs[B*N*2] (interleaved cos,sin) | CS[B*2] | sums[B*3*D]
    float* ws   = (float*)d_ws;
    float* cs   = ws;
    float* CS   = cs + B_ * N_ * 2;
    float* sums = CS + B_ * 2;

    const int nzero = B_ * 2 + B_ * 3 * D_;        // CS and sums are contiguous
    pwa_zero<<<(nzero + 255) / 256, 256, 0, stream>>>(CS, nzero);
    pwa_phase<<<(B_ * N_) / 256, 256, 0, stream>>>(phases, cs, CS);
    pwa_wmma_reduce<<<B_ * NT_ * NC_, 32, 0, stream>>>(hidden, cs, sums);
    pwa_finalize<<<B_ * (N_ / 16), 256, 0, stream>>>(hidden, cs, CS, sums, alpha, out);
}